// QTAttAPytorch_39633958207874
// MI455X (gfx1250) — compile-verified
//
#include <hip/hip_runtime.h>
#include <hip/hip_bf16.h>

// QTAttA quadtree attention, 3 levels (coarse 16x16 -> 32x32 -> 64x64).
// B=4, H=8, D=32, K(topk)=8.
// Coarse-level GEMMs: V_WMMA_F32_16X16X4_F32, with K/V operands staged in LDS
// via async global->LDS copies (ASYNCcnt) when the toolchain exposes them.

typedef float v2f __attribute__((ext_vector_type(2)));
typedef float v8f __attribute__((ext_vector_type(8)));

#define NB    4
#define NHEAD 8
#define DIMH  32
#define TOPK  8
#define SC    256           // coarse tokens (16x16)
#define TEMP  0.17677669529663687f   // 1/sqrt(32)

#if defined(__has_builtin)
#if __has_builtin(__builtin_amdgcn_global_load_async_to_lds_b32)
#define HAVE_ASYNC_LDS 1
#endif
#endif

typedef __attribute__((address_space(1))) int* as1ip;
typedef __attribute__((address_space(3))) int* as3ip;

__device__ __forceinline__ void ld_to_lds_f32(float* lds_dst, const float* gsrc) {
#ifdef HAVE_ASYNC_LDS
  float* g = const_cast<float*>(gsrc);
  __builtin_amdgcn_global_load_async_to_lds_b32((as1ip)g, (as3ip)lds_dst, 0, 0);
#else
  *lds_dst = *gsrc;
#endif
}

__device__ __forceinline__ void wait_lds_copies() {
#ifdef HAVE_ASYNC_LDS
#if __has_builtin(__builtin_amdgcn_s_wait_asynccnt)
  __builtin_amdgcn_s_wait_asynccnt(0);
#else
  asm volatile("s_wait_asynccnt 0x0" ::: "memory");
#endif
#endif
}

// ---------------------------------------------------------------------------
// Kernel A: coarse QK^T.  A0[bh][l][s] = sum_d Q[l][d] * K[s][d]
// One block (8 waves) per (b,head).  K block (32x256 f32 = 32KB, contiguous)
// staged once in LDS asynchronously; each wave computes 32 of the 256
// 16x16 output tiles with WMMA f32 16x16x4 over D=32 (8 steps).
// ---------------------------------------------------------------------------
__global__ __launch_bounds__(256) void qk_coarse(const float* __restrict__ q,
                                                 const float* __restrict__ k,
                                                 float* __restrict__ A0) {
  __shared__ float sk[DIMH * SC];                  // [d][s], 32 KB
  const int bh = blockIdx.x;                       // b*8 + head
  const int b = bh >> 3, head = bh & 7;
  const float* qb = q + ((size_t)b * 256 + head * DIMH) * SC;  // [d][l]
  const float* kb = k + ((size_t)b * 256 + head * DIMH) * SC;  // [d][s] contiguous

  for (int i = threadIdx.x; i < DIMH * SC; i += 256)
    ld_to_lds_f32(&sk[i], &kb[i]);
  wait_lds_copies();
  __syncthreads();

  const int wave = threadIdx.x >> 5;
  const int lane = threadIdx.x & 31;
  const int half = lane >> 4;
  const int lm = lane & 15;
  float* out = A0 + (size_t)bh * SC * SC;

  for (int it = 0; it < 32; ++it) {
    const int t = (it << 3) | wave;                // tile 0..255
    const int l0 = (t >> 4) << 4;
    const int s0 = (t & 15) << 4;
    v8f acc = {0.f,0.f,0.f,0.f,0.f,0.f,0.f,0.f};
#pragma unroll
    for (int dk = 0; dk < DIMH; dk += 4) {
      v2f a, bm;
      a.x  = qb[(size_t)(dk + 2 * half + 0) * SC + l0 + lm];
      a.y  = qb[(size_t)(dk + 2 * half + 1) * SC + l0 + lm];
      bm.x = sk[(dk + 2 * half + 0) * SC + s0 + lm];
      bm.y = sk[(dk + 2 * half + 1) * SC + s0 + lm];
      acc = __builtin_amdgcn_wmma_f32_16x16x4_f32(false, a, false, bm,
                                                  (short)0, acc, false, false);
    }
#pragma unroll
    for (int i = 0; i < 8; ++i)
      out[(size_t)(l0 + i + 8 * half) * SC + s0 + lm] = acc[i];
  }
}

// ---------------------------------------------------------------------------
// Kernel B: per-row softmax(temp*QK) over s, then top-8 (stable, low index
// wins ties, matching jax.lax.top_k).  Normalized A written back in place.
// One thread per (b, head, l):  4*8*256 = 8192 threads.
// ---------------------------------------------------------------------------
__global__ __launch_bounds__(256) void coarse_softmax_topk(float* __restrict__ A0,
                                                           float* __restrict__ sc0,
                                                           int* __restrict__ ix0) {
  const int t = blockIdx.x * 256 + threadIdx.x;
  const int bh = t >> 8;
  const int l = t & 255;
  const int b = bh >> 3, head = bh & 7;
  float* row = A0 + ((size_t)bh * SC + l) * SC;
  float mx = -3.4e38f;
  for (int s = 0; s < SC; ++s) mx = fmaxf(mx, row[s]);
  mx *= TEMP;
  float sum = 0.f;
  for (int s = 0; s < SC; ++s) {
    float e = __expf(TEMP * row[s] - mx);
    row[s] = e;
    sum += e;
  }
  const float inv = 1.f / sum;
  for (int s = 0; s < SC; ++s) row[s] *= inv;

  float bs[TOPK]; int bi[TOPK];
#pragma unroll
  for (int kk = 0; kk < TOPK; ++kk) {
    float bv = -1.f; int bj = 0;
    for (int s = 0; s < SC; ++s) {
      float v = row[s];
      bool taken = false;
#pragma unroll
      for (int j = 0; j < TOPK; ++j) taken |= (j < kk) && (bi[j] == s);
      if (!taken && v > bv) { bv = v; bj = s; }
    }
    bs[kk] = bv; bi[kk] = bj;
  }
#pragma unroll
  for (int kk = 0; kk < TOPK; ++kk) {
    sc0[((size_t)(b * SC + l) * TOPK + kk) * NHEAD + head] = bs[kk];
    ix0[((size_t)(b * SC + l) * TOPK + kk) * NHEAD + head] = bi[kk];
  }
}

// ---------------------------------------------------------------------------
// Kernel C: msg0 = A0 * V  ([256x256] x [256x32] per (b,head)) via WMMA.
// One block per (b,head); V (32KB) staged once in LDS asynchronously.
// msg0 layout: [b, l, H, D].
// ---------------------------------------------------------------------------
__global__ __launch_bounds__(256) void av_coarse(const float* __restrict__ A0,
                                                 const float* __restrict__ v,
                                                 float* __restrict__ msg0) {
  __shared__ float sv[DIMH * SC];                  // [d][s], 32 KB
  const int bh = blockIdx.x;
  const int b = bh >> 3, head = bh & 7;
  const float* vb = v + ((size_t)b * 256 + head * DIMH) * SC;  // contiguous

  for (int i = threadIdx.x; i < DIMH * SC; i += 256)
    ld_to_lds_f32(&sv[i], &vb[i]);
  wait_lds_copies();
  __syncthreads();

  const int wave = threadIdx.x >> 5;
  const int lane = threadIdx.x & 31;
  const int half = lane >> 4;
  const int lm = lane & 15;
  const float* Ar = A0 + (size_t)bh * SC * SC;     // [l][s]

  for (int it = 0; it < 4; ++it) {
    const int t = (it << 3) | wave;                // tile 0..31
    const int l0 = (t >> 1) << 4;
    const int d0 = (t & 1) << 4;
    v8f acc = {0.f,0.f,0.f,0.f,0.f,0.f,0.f,0.f};
    for (int k0 = 0; k0 < SC; k0 += 4) {
      v2f a, bm;
      a.x  = Ar[(size_t)(l0 + lm) * SC + k0 + 2 * half + 0];
      a.y  = Ar[(size_t)(l0 + lm) * SC + k0 + 2 * half + 1];
      bm.x = sv[(d0 + lm) * SC + k0 + 2 * half + 0];
      bm.y = sv[(d0 + lm) * SC + k0 + 2 * half + 1];
      acc = __builtin_amdgcn_wmma_f32_16x16x4_f32(false, a, false, bm,
                                                  (short)0, acc, false, false);
    }
#pragma unroll
    for (int i = 0; i < 8; ++i) {
      int l = l0 + i + 8 * half;
      msg0[((size_t)(b * SC + l) * NHEAD + head) * DIMH + d0 + lm] = acc[i];
    }
  }
}

// ---------------------------------------------------------------------------
// Kernel D: subtract top-k value contributions (mask term) from msg0.
// One thread per (b, l, head).
// ---------------------------------------------------------------------------
__global__ __launch_bounds__(256) void coarse_sub_topk(float* __restrict__ msg0,
                                                       const float* __restrict__ sc0,
                                                       const int* __restrict__ ix0,
                                                       const float* __restrict__ v) {
  const int t = blockIdx.x * 256 + threadIdx.x;   // (b*256+l)*8+head
  const int head = t & 7;
  const int bl = t >> 3;
  const int l = bl & 255;
  const int b = bl >> 8;
  const float* vb = v + ((size_t)b * 256 + head * DIMH) * SC;
  float* m = msg0 + ((size_t)(b * SC + l) * NHEAD + head) * DIMH;
  float acc[DIMH];
#pragma unroll
  for (int d = 0; d < DIMH; ++d) acc[d] = m[d];
#pragma unroll
  for (int kk = 0; kk < TOPK; ++kk) {
    float s = sc0[((size_t)(b * SC + l) * TOPK + kk) * NHEAD + head];
    int ix = ix0[((size_t)(b * SC + l) * TOPK + kk) * NHEAD + head];
#pragma unroll
    for (int d = 0; d < DIMH; ++d) acc[d] -= s * vb[(size_t)d * SC + ix];
  }
#pragma unroll
  for (int d = 0; d < DIMH; ++d) m[d] = acc[d];
}

// ---------------------------------------------------------------------------
// Kernel E: one refinement level.  One thread per (b, parent cell l, head).
// Gathers 8 parents * 4 children = 32 keys, softmax over 4-child groups,
// weights by parent score, accumulates message, optional top-8 + mask,
// scatters to fine-token order [b, w*w, H, D].
// ---------------------------------------------------------------------------
__global__ __launch_bounds__(256) void refine_level(
    const float* __restrict__ q, const float* __restrict__ k,
    const float* __restrict__ v,
    const float* __restrict__ msg_in, const float* __restrict__ sc_in,
    const int* __restrict__ ix_in,
    float* __restrict__ msg_out, float* __restrict__ sc_out,
    int* __restrict__ ix_out,
    int w, int wprev, int Lp, int S, int has_topk) {
  const int tid = blockIdx.x * 256 + threadIdx.x;
  if (tid >= NB * Lp * NHEAD) return;
  const int head = tid & 7;
  const int bl = tid >> 3;
  const int l = bl % Lp;
  const int b = bl / Lp;
  const int w2 = w >> 1;
  const int y2 = l / w2, x2 = l % w2;

  const float* qb = q + ((size_t)b * 256 + head * DIMH) * S;  // [d][token]
  const float* kb = k + ((size_t)b * 256 + head * DIMH) * S;
  const float* vb = v + ((size_t)b * 256 + head * DIMH) * S;

  int gidx[4 * TOPK];
  float score[TOPK];
#pragma unroll
  for (int kk = 0; kk < TOPK; ++kk) {
    int pidx = ix_in[((size_t)(b * Lp + l) * TOPK + kk) * NHEAD + head];
    score[kk] = sc_in[((size_t)(b * Lp + l) * TOPK + kk) * NHEAD + head];
    int r = (pidx / wprev) * 2;
    int c = (pidx % wprev) * 2;
    gidx[kk * 4 + 0] = r * w + c;
    gidx[kk * 4 + 1] = r * w + c + 1;
    gidx[kk * 4 + 2] = (r + 1) * w + c;
    gidx[kk * 4 + 3] = (r + 1) * w + c + 1;
  }

  const float* pm = msg_in + ((size_t)(b * Lp + l) * NHEAD + head) * DIMH;

  for (int wc = 0; wc < 4; ++wc) {
    const int t1 = wc >> 1, t2 = wc & 1;
    const int tok = (2 * y2 + t1) * w + (2 * x2 + t2);
    float qv[DIMH], msg[DIMH], A[4 * TOPK];
#pragma unroll
    for (int d = 0; d < DIMH; ++d) qv[d] = qb[(size_t)d * S + tok];
#pragma unroll
    for (int d = 0; d < DIMH; ++d) msg[d] = pm[d];

#pragma unroll
    for (int kk = 0; kk < TOPK; ++kk) {
      float qk4[4];
#pragma unroll
      for (int f = 0; f < 4; ++f) {
        int s = gidx[kk * 4 + f];
        float acc = 0.f;
#pragma unroll
        for (int d = 0; d < DIMH; ++d) acc += qv[d] * kb[(size_t)d * S + s];
        qk4[f] = acc * TEMP;
      }
      float m = fmaxf(fmaxf(qk4[0], qk4[1]), fmaxf(qk4[2], qk4[3]));
      float e0 = __expf(qk4[0] - m), e1 = __expf(qk4[1] - m);
      float e2 = __expf(qk4[2] - m), e3 = __expf(qk4[3] - m);
      float inv = score[kk] / (e0 + e1 + e2 + e3);
      A[kk * 4 + 0] = e0 * inv;
      A[kk * 4 + 1] = e1 * inv;
      A[kk * 4 + 2] = e2 * inv;
      A[kk * 4 + 3] = e3 * inv;
    }

#pragma unroll
    for (int kf = 0; kf < 4 * TOPK; ++kf) {
      int s = gidx[kf];
      float wgt = A[kf];
#pragma unroll
      for (int d = 0; d < DIMH; ++d) msg[d] += wgt * vb[(size_t)d * S + s];
    }

    if (has_topk) {
      float tmp[4 * TOPK];
#pragma unroll
      for (int i = 0; i < 4 * TOPK; ++i) tmp[i] = A[i];
      float bs[TOPK]; int bi[TOPK];
#pragma unroll
      for (int kk = 0; kk < TOPK; ++kk) {
        float bv = -3.4e38f; int bj = 0;
#pragma unroll
        for (int i = 0; i < 4 * TOPK; ++i)
          if (tmp[i] > bv) { bv = tmp[i]; bj = i; }
        bs[kk] = bv; bi[kk] = bj; tmp[bj] = -3.4e38f;
      }
#pragma unroll
      for (int kk = 0; kk < TOPK; ++kk) {
        int s = gidx[bi[kk]];
        float sv = bs[kk];
#pragma unroll
        for (int d = 0; d < DIMH; ++d) msg[d] -= sv * vb[(size_t)d * S + s];
      }
#pragma unroll
      for (int kk = 0; kk < TOPK; ++kk) {
        sc_out[((size_t)(b * S + tok) * TOPK + kk) * NHEAD + head] = bs[kk];
        ix_out[((size_t)(b * S + tok) * TOPK + kk) * NHEAD + head] = gidx[bi[kk]];
      }
    }

#pragma unroll
    for (int d = 0; d < DIMH; ++d)
      msg_out[((size_t)(b * S + tok) * NHEAD + head) * DIMH + d] = msg[d];
  }
}

// ---------------------------------------------------------------------------
// Host launcher
// ---------------------------------------------------------------------------
extern "C" void kernel_launch(void* const* d_in, const int* in_sizes, int n_in,
                              void* d_out, int out_size, void* d_ws, size_t ws_size,
                              hipStream_t stream) {
  // setup_inputs order: queries_0..2, keys_0..2, values_0..2 (lvl0=64 fine, lvl2=16 coarse)
  const float* q0 = (const float*)d_in[0];  // 64x64
  const float* q1 = (const float*)d_in[1];  // 32x32
  const float* q2 = (const float*)d_in[2];  // 16x16 (coarse)
  const float* k0 = (const float*)d_in[3];
  const float* k1 = (const float*)d_in[4];
  const float* k2 = (const float*)d_in[5];
  const float* v0 = (const float*)d_in[6];
  const float* v1 = (const float*)d_in[7];
  const float* v2 = (const float*)d_in[8];

  float* ws = (float*)d_ws;
  float* A0   = ws;                                  // 4*8*256*256 = 2,097,152 f
  float* msg0 = A0 + (size_t)NB * NHEAD * SC * SC;   // 4*256*8*32  =   262,144 f
  float* sc0  = msg0 + (size_t)NB * SC * NHEAD * DIMH;       // 65,536 f
  int*   ix0  = (int*)(sc0 + (size_t)NB * SC * TOPK * NHEAD);// 65,536 i
  float* msg1 = (float*)(ix0 + (size_t)NB * SC * TOPK * NHEAD); // 4*1024*256 = 1,048,576 f
  float* sc1  = msg1 + (size_t)NB * 1024 * NHEAD * DIMH;        // 262,144 f
  int*   ix1  = (int*)(sc1 + (size_t)NB * 1024 * TOPK * NHEAD); // 262,144 i

  // Level 0 (coarse 16x16): dense attention via WMMA + async K/V staging
  qk_coarse<<<NB * NHEAD, 256, 0, stream>>>(q2, k2, A0);
  coarse_softmax_topk<<<(NB * NHEAD * SC) / 256, 256, 0, stream>>>(A0, sc0, ix0);
  av_coarse<<<NB * NHEAD, 256, 0, stream>>>(A0, v2, msg0);
  coarse_sub_topk<<<(NB * SC * NHEAD) / 256, 256, 0, stream>>>(msg0, sc0, ix0, v2);

  // Level 1 (32x32): refine, produces msg1 + topk for level 2
  refine_level<<<(NB * 256 * NHEAD) / 256, 256, 0, stream>>>(
      q1, k1, v1, msg0, sc0, ix0, msg1, sc1, ix1,
      /*w=*/32, /*wprev=*/16, /*Lp=*/256, /*S=*/1024, /*has_topk=*/1);

  // Level 2 (64x64): final refine, writes d_out [b, 4096, 8, 32]
  refine_level<<<(NB * 1024 * NHEAD) / 256, 256, 0, stream>>>(
      q0, k0, v0, msg1, sc1, ix1, (float*)d_out, nullptr, nullptr,
      /*w=*/64, /*wprev=*/32, /*Lp=*/1024, /*S=*/4096, /*has_topk=*/0);
}